// Framework_31379031065134
// MI455X (gfx1250) — compile-verified
//
#include <hip/hip_runtime.h>
#include <cmath>

#define S  64
#define F  10
#define K  7
#define D  512
#define HH 14
#define DT 128
#define HW (HH*HH)   // 196
#define NV (S*F)     // 640

typedef float v2f __attribute__((ext_vector_type(2)));
typedef float v8f __attribute__((ext_vector_type(8)));
typedef int   v4i __attribute__((vector_size(16)));   // matches async-builtin param

#if defined(__has_builtin)
#if __has_builtin(__builtin_amdgcn_global_load_async_to_lds_b128)
#define HAVE_ASYNC_LDS 1
#endif
#endif

// feat LDS row stride: 204 floats = 816B -> 16B aligned rows (B128 async),
// 204 % 64 = 12, gcd(12,64)=4 -> 16 consecutive rows hit 16 distinct banks.
#define FSTRIDE 204

// ---------------------------------------------------------------------------
// Kernel 1: fg[n,k,c] = sum_hw feat[n,c,hw]*cam[n,k,hw] / (sum_hw cam[n,k,hw]+1e-10)
// Bandwidth-bound (streams 257MB of feat once). WMMA f32 16x16x4 does the
// contraction: A = cam (M=k rows, zero-padded to 16), B = feat^T (hw x c).
// grid: (NV, D/64), block: 128 threads (4 waves, each wave owns 16 channels)
// ---------------------------------------------------------------------------
__global__ __launch_bounds__(128) void einsum_kernel(
    const float* __restrict__ feat,  // (NV, D, HW)
    const float* __restrict__ cam,   // (NV, K, HW)
    float* __restrict__ fg)          // (NV, K, D)
{
  __shared__ float sCam[16*HW];         // 12544 B, rows 7..15 zero
  __shared__ float sInv[8];
  __shared__ float sFeat[64*FSTRIDE];   // 52224 B

  const int n   = blockIdx.x;
  const int c0  = blockIdx.y * 64;
  const int tid = threadIdx.x;

  // stage cam[n] (7x196) and zero-pad rows 7..15 (kills EXEC masking in loop)
  for (int i = tid; i < K*HW; i += 128)  sCam[i] = cam[(size_t)n*K*HW + i];
  for (int i = K*HW + tid; i < 16*HW; i += 128) sCam[i] = 0.f;

  // stage feat tile (64 rows x 196): CDNA5 async global->LDS when available
  for (int i = tid; i < 64*49; i += 128) {
    int row = i / 49, col4 = i % 49;
    const float* src = feat + ((size_t)n*D + c0 + row)*HW + col4*4;
    float* dst = &sFeat[row*FSTRIDE + col4*4];
#if HAVE_ASYNC_LDS
    __builtin_amdgcn_global_load_async_to_lds_b128((v4i*)src, (v4i*)dst, 0, 0);
#else
    const float4 v = *reinterpret_cast<const float4*>(src);
    dst[0]=v.x; dst[1]=v.y; dst[2]=v.z; dst[3]=v.w;
#endif
  }
#if HAVE_ASYNC_LDS
  asm volatile("s_wait_asynccnt 0" ::: "memory");
#endif
  __syncthreads();
  if (tid < K) {
    float s = 0.f;
    for (int j = 0; j < HW; ++j) s += sCam[tid*HW + j];
    sInv[tid] = 1.0f / (s + 1e-10f);
  }
  __syncthreads();

  const int wave = tid >> 5;
  const int lane = tid & 31;
  const int half = lane >> 4;        // which 16-lane half
  const int l16  = lane & 15;
  const int kb   = half * 2;         // K offset within the group of 4

  const int m    = l16;              // A-matrix row (k index, rows >=7 are 0)
  const int crow = wave*16 + l16;    // B-matrix column (channel)

  v8f acc = {};
  for (int kk = 0; kk < HW; kk += 4) {
    v2f a, b;
    a.x = sCam[m*HW + kk + kb];
    a.y = sCam[m*HW + kk + kb + 1];
    b.x = sFeat[crow*FSTRIDE + kk + kb];
    b.y = sFeat[crow*FSTRIDE + kk + kb + 1];
    acc = __builtin_amdgcn_wmma_f32_16x16x4_f32(false, a, false, b,
                                                (short)0, acc, false, false);
  }

  // C layout: VGPR r -> M = r + half*8, N = l16
  #pragma unroll
  for (int r = 0; r < 8; ++r) {
    int mo = r + half*8;
    if (mo < K)
      fg[((size_t)n*K + mo)*D + c0 + wave*16 + l16] = acc[r] * sInv[mo];
  }
}

// ---------------------------------------------------------------------------
// Kernel 2: T = ((relu(X@W1 + b1)) @ W2 + b2) * g + bn      (f32 WMMA 16x16x4)
// grid: R/16 blocks, block: 256 threads (8 waves; wave w owns output cols 16w..)
// ---------------------------------------------------------------------------
__global__ __launch_bounds__(256) void mlp_kernel(
    const float* __restrict__ X,    // (R, 512)
    const float* __restrict__ W1,   // (512, 128)
    const float* __restrict__ b1,   // (128)
    const float* __restrict__ W2,   // (128, 128)
    const float* __restrict__ b2,   // (128)
    const float* __restrict__ g,    // (128)
    const float* __restrict__ bn,   // (128)
    float* __restrict__ T)          // (R, 128)
{
  __shared__ float sX[16*516];      // 33024 B (stride pad 512->516)
  __shared__ float sH[16*132];      // 8448 B  (stride pad 128->132)

  const int row0 = blockIdx.x * 16;
  const int tid  = threadIdx.x;

  // stage 16x512 X rows, coalesced float4
  for (int i = tid; i < 16*128; i += 256) {
    int r = i >> 7, c4 = i & 127;
    float4 v = reinterpret_cast<const float4*>(X + (size_t)(row0 + r)*D)[c4];
    float* dst = &sX[r*516 + c4*4];
    dst[0]=v.x; dst[1]=v.y; dst[2]=v.z; dst[3]=v.w;
  }
  __syncthreads();

  const int wave = tid >> 5, lane = tid & 31;
  const int half = lane >> 4, l16 = lane & 15;
  const int kb   = half * 2;
  const int col0 = wave * 16;

  // GEMM1: H = relu(X @ W1 + b1)
  v8f acc = {};
  for (int kk = 0; kk < D; kk += 4) {
    v2f a, b;
    a.x = sX[l16*516 + kk + kb];
    a.y = sX[l16*516 + kk + kb + 1];
    b.x = W1[(size_t)(kk + kb    )*DT + col0 + l16];
    b.y = W1[(size_t)(kk + kb + 1)*DT + col0 + l16];
    acc = __builtin_amdgcn_wmma_f32_16x16x4_f32(false, a, false, b,
                                                (short)0, acc, false, false);
  }
  {
    float bias = b1[col0 + l16];
    #pragma unroll
    for (int r = 0; r < 8; ++r) {
      int mo = r + half*8;
      float h = acc[r] + bias;
      sH[mo*132 + col0 + l16] = h > 0.f ? h : 0.f;
    }
  }
  __syncthreads();

  // GEMM2: Y = H @ W2 + b2 ; T = Y*g + bn
  v8f acc2 = {};
  for (int kk = 0; kk < DT; kk += 4) {
    v2f a, b;
    a.x = sH[l16*132 + kk + kb];
    a.y = sH[l16*132 + kk + kb + 1];
    b.x = W2[(size_t)(kk + kb    )*DT + col0 + l16];
    b.y = W2[(size_t)(kk + kb + 1)*DT + col0 + l16];
    acc2 = __builtin_amdgcn_wmma_f32_16x16x4_f32(false, a, false, b,
                                                 (short)0, acc2, false, false);
  }
  {
    float bias = b2[col0 + l16], gg = g[col0 + l16], bb = bn[col0 + l16];
    #pragma unroll
    for (int r = 0; r < 8; ++r) {
      int mo = r + half*8;
      T[(size_t)(row0 + mo)*DT + col0 + l16] = (acc2[r] + bias) * gg + bb;
    }
  }
}

// ---------------------------------------------------------------------------
// Kernel 3: losses + masks. One wave32 per (s,k,f) cell; 128-dim reduction.
// out layout: [loss_co*m_co | loss_di*m_di | mask_co | mask_di], each S*K*F.
// ---------------------------------------------------------------------------
__global__ __launch_bounds__(256) void loss_kernel(
    const float* __restrict__ Ta,       // (S*K, 128)
    const float* __restrict__ Tv,       // (S*F*K, 128)
    const float* __restrict__ pred_a,   // (S, K)
    const float* __restrict__ pred_v,   // (S*F, K)
    const int*   __restrict__ perm_idx, // (S, K, F)
    const int*   __restrict__ cls_idx,  // (S, K, F)
    float* __restrict__ out)
{
  const int wid  = (blockIdx.x * blockDim.x + threadIdx.x) >> 5;
  const int lane = threadIdx.x & 31;
  if (wid >= S*K*F) return;
  const int f = wid % F;
  const int k = (wid / F) % K;
  const int s = wid / (F*K);

  const float* ta = Ta + (size_t)(s*K + k)*DT;
  const float* tv = Tv + (size_t)((s*F + f)*K + k)*DT;
  const int s2 = s ^ 1;
  const int ff = perm_idx[(s*K + k)*F + f];
  const int k2 = cls_idx [(s*K + k)*F + f];
  const float* tvd = Tv + (size_t)((s2*F + ff)*K + k2)*DT;

  float sco = 0.f, sdi = 0.f;
  #pragma unroll
  for (int d = lane; d < DT; d += 32) {
    float tav = ta[d];
    float d1 = tav - tv[d];
    float d2 = tav - tvd[d];
    sco += d1*d1;
    sdi += d2*d2;
  }
  #pragma unroll
  for (int o = 16; o > 0; o >>= 1) {
    sco += __shfl_down(sco, o, 32);
    sdi += __shfl_down(sdi, o, 32);
  }
  if (lane == 0) {
    float pa  = pred_a[s*K + k];
    bool  act = pa > 0.3f;
    float pv  = 1.0f / (1.0f + __expf(-pred_v[(s*F + f)*K + k]));
    bool  mco = act && (pv > 0.3f);
    int   num = (int)floorf(pa * (float)F);
    bool  mdi = act && (f < num);
    const int idx = (s*K + k)*F + f;
    const int N   = S*K*F;
    out[idx]       = mco ? sco * (1.0f/DT) : 0.f;
    out[N + idx]   = mdi ? sdi * (1.0f/DT) : 0.f;
    out[2*N + idx] = mco ? 1.f : 0.f;
    out[3*N + idx] = mdi ? 1.f : 0.f;
  }
}

// ---------------------------------------------------------------------------
extern "C" void kernel_launch(void* const* d_in, const int* in_sizes, int n_in,
                              void* d_out, int out_size, void* d_ws, size_t ws_size,
                              hipStream_t stream) {
  const float* feat_a  = (const float*)d_in[0];
  const float* pred_a  = (const float*)d_in[1];
  const float* feat_v  = (const float*)d_in[2];
  const float* pred_v  = (const float*)d_in[3];
  const float* cam     = (const float*)d_in[4];
  const float* W1a = (const float*)d_in[5];
  const float* b1a = (const float*)d_in[6];
  const float* W2a = (const float*)d_in[7];
  const float* b2a = (const float*)d_in[8];
  const float* ga  = (const float*)d_in[9];
  const float* bna = (const float*)d_in[10];
  const float* W1v = (const float*)d_in[11];
  const float* b1v = (const float*)d_in[12];
  const float* W2v = (const float*)d_in[13];
  const float* b2v = (const float*)d_in[14];
  const float* gv  = (const float*)d_in[15];
  const float* bnv = (const float*)d_in[16];
  const int* perm_idx = (const int*)d_in[17];
  const int* cls_idx  = (const int*)d_in[18];

  float* ws = (float*)d_ws;
  float* fg = ws;                           // NV*K*D     = 2,293,760 floats
  float* Ta = fg + (size_t)NV*K*D;          // S*K*DT     = 57,344
  float* Tv = Ta + (size_t)S*K*DT;          // S*F*K*DT   = 573,440

  einsum_kernel<<<dim3(NV, D/64), 128, 0, stream>>>(feat_v, cam, fg);
  mlp_kernel<<<(S*K)/16,   256, 0, stream>>>(feat_a, W1a, b1a, W2a, b2a, ga, bna, Ta);
  mlp_kernel<<<(S*F*K)/16, 256, 0, stream>>>(fg,     W1v, b1v, W2v, b2v, gv, bnv, Tv);
  loss_kernel<<<(S*K*F)/8, 256, 0, stream>>>(Ta, Tv, pred_a, pred_v,
                                             perm_idx, cls_idx, (float*)d_out);
}